// skip_gram_1236950581668
// MI455X (gfx1250) — compile-verified
//
#include <hip/hip_runtime.h>
#include <math.h>

// Problem constants (fixed by the reference)
#define VOCAB 50000
#define D 128
#define BATCH 128
#define CTX 10
#define NEG 5

// LDS strides (in dwords) chosen for bank-conflict-free B-fragment reads:
//  - Vpos is read column-wise (s1 B-matrix): stride 136 -> rows k and k+2
//    differ by 272 dwords = +16 banks, disjoint from the 16-lane column span.
//  - Vneg is read row-wise as b64 (s2 B-matrix): stride 132 -> 16 distinct
//    rows hit banks {0,4,8,...,60}, all distinct.
#define VP_STRIDE 136
#define VN_STRIDE 132

typedef float v2f __attribute__((ext_vector_type(2)));
typedef float v8f __attribute__((ext_vector_type(8)));

__device__ __forceinline__ float softplus_stable(float v) {
    // softplus(v) = log(1 + exp(v)) = max(v,0) + log1p(exp(-|v|))
    float av = fabsf(v);
    return fmaxf(v, 0.0f) + log1pf(expf(-av));
}

__global__ __launch_bounds__(256, 1)
void skip_gram_loss_kernel(const int* __restrict__ x,      // [128,10]
                           const int* __restrict__ label,  // [128]
                           const int* __restrict__ negs,   // [128,5]
                           const float* __restrict__ emb,  // [50000,128]
                           float* __restrict__ out)        // [1]
{
    __shared__ float sVpos[BATCH * VP_STRIDE];   // ~68 KB
    __shared__ float sVneg[BATCH * VN_STRIDE];   // ~66 KB
    __shared__ float sWaveSum[8];

    const int tid = threadIdx.x;

    // ---------------- Phase 0: gather + mean into LDS (coalesced rows) -----
    for (int idx = tid; idx < BATCH * D; idx += 256) {
        const int k = idx >> 7;          // batch row
        const int j = idx & 127;         // embed dim
        float s = 0.0f;
        #pragma unroll
        for (int c = 0; c < CTX; ++c)
            s += emb[(size_t)x[k * CTX + c] * D + j];
        sVpos[k * VP_STRIDE + j] = s * (1.0f / CTX);
    }
    for (int idx = tid; idx < BATCH * D; idx += 256) {
        const int k = idx >> 7;
        const int j = idx & 127;
        float s = 0.0f;
        #pragma unroll
        for (int n = 0; n < NEG; ++n)
            s += emb[(size_t)negs[k * NEG + n] * D + j];
        sVneg[k * VN_STRIDE + j] = s * (1.0f / NEG);
    }
    __syncthreads();

    // ---------------- Phase 1: diagonal-tile WMMA (fp32, 16x16x4) ----------
    // Wave t owns diagonal tile t: rows/cols [16t, 16t+16).
    const int wave = tid >> 5;
    const int lane = tid & 31;
    const int half = lane >> 4;      // 0: K = 4s+{0,1}, 1: K = 4s+{2,3}
    const int lid  = lane & 15;      // M (for A) / N (for B)
    const int t0   = wave * 16;
    const int col  = t0 + lid;

    // A-matrix row for this lane: u[t0+lid, :] = emb[label[t0+lid], :]
    const int arow = label[t0 + lid];
    const float* aptr = emb + (size_t)arow * D + 2 * half;
    __builtin_prefetch(aptr, 0, 0);   // global_prefetch_b8: warm the U row

    v8f acc1 = {};   // tile of U @ Vpos        -> diag = s1
    v8f acc2 = {};   // tile of U @ Vneg^T      -> diag = s2

    #pragma unroll
    for (int s = 0; s < 32; ++s) {
        const int k0 = 4 * s + 2 * half;
        // A frag: lane<16 holds {K=4s, 4s+1}, lane>=16 holds {K=4s+2, 4s+3}
        v2f a = *(const v2f*)(aptr + 4 * s);
        // B1 frag: B1[k][n] = Vpos[k][t0+n]  (column slice, strided rows)
        v2f b1;
        b1.x = sVpos[(k0    ) * VP_STRIDE + col];
        b1.y = sVpos[(k0 + 1) * VP_STRIDE + col];
        // B2 frag: B2[k][n] = Vneg[t0+n][k]  (row slice, contiguous -> b64)
        v2f b2 = *(const v2f*)(&sVneg[col * VN_STRIDE + k0]);

        acc1 = __builtin_amdgcn_wmma_f32_16x16x4_f32(
            false, a, false, b1, (short)0, acc1, false, false);
        acc2 = __builtin_amdgcn_wmma_f32_16x16x4_f32(
            false, a, false, b2, (short)0, acc2, false, false);
    }

    // ---------------- Phase 2: diagonal extract + loss + reduction ----------
    // f32 C/D layout: lanes 0-15: M=vgpr, N=lane; lanes 16-31: M=vgpr+8, N=lane-16.
    // Diagonal (M==N): lanes 0-7 at vgpr=lane, lanes 24-31 at vgpr=lane-24.
    float lane_val = 0.0f;
    if (lane < 8 || lane >= 24) {
        const int vi = (lane < 8) ? lane : (lane - 24);
        const float s1 = acc1[vi];
        const float s2 = acc2[vi];
        // -log_sigmoid(s1) = softplus(-s1);  -log_sigmoid(-s2) = softplus(s2)
        lane_val = softplus_stable(-s1) + softplus_stable(s2);
    }
    #pragma unroll
    for (int off = 16; off >= 1; off >>= 1)
        lane_val += __shfl_xor(lane_val, off, 32);
    if (lane == 0) sWaveSum[wave] = lane_val;
    __syncthreads();

    if (tid == 0) {
        float acc = 0.0f;
        #pragma unroll
        for (int w = 0; w < 8; ++w) acc += sWaveSum[w];   // fixed order: deterministic
        out[0] = acc * (1.0f / (float)BATCH);
    }
}

extern "C" void kernel_launch(void* const* d_in, const int* in_sizes, int n_in,
                              void* d_out, int out_size, void* d_ws, size_t ws_size,
                              hipStream_t stream) {
    const int*   x     = (const int*)d_in[0];    // [128,10]
    const int*   label = (const int*)d_in[1];    // [128]
    const int*   negs  = (const int*)d_in[2];    // [128,5]
    const float* emb   = (const float*)d_in[3];  // [50000,128]
    float*       out   = (float*)d_out;          // [1]

    hipLaunchKernelGGL(skip_gram_loss_kernel, dim3(1), dim3(256), 0, stream,
                       x, label, negs, emb, out);
}